// GCNNet_6012954215114
// MI455X (gfx1250) — compile-verified
//
#include <hip/hip_runtime.h>

typedef __attribute__((ext_vector_type(2))) float v2f;
typedef __attribute__((ext_vector_type(8))) float v8f;

__device__ __forceinline__ v8f wmma_f32_16x16x4(v2f a, v2f b, v8f c) {
  // D = A(16x4) * B(4x16) + C(16x16), fp32, wave32
  return __builtin_amdgcn_wmma_f32_16x16x4_f32(false, a, false, b, (short)0, c,
                                               false, false);
}

// ---------------- prep: zero-pad small weight matrices ----------------
// w1p: [8][32] from W1 [5][32]; w3p: [16][16] from W3 [16][3]; b3p: [4] from b3 [3]
__global__ void k_prep(const float* __restrict__ W1, const float* __restrict__ W3,
                       const float* __restrict__ b3, float* __restrict__ w1p,
                       float* __restrict__ w3p, float* __restrict__ b3p) {
  int t = threadIdx.x;  // 256 threads, 1 block
  { int r = t >> 5, c = t & 31; w1p[t] = (r < 5) ? W1[r * 32 + c] : 0.f; }
  { int r = t >> 4, c = t & 15; w3p[t] = (c < 3) ? W3[r * 3 + c] : 0.f; }
  if (t < 4) b3p[t] = (t < 3) ? b3[t] : 0.f;
}

// ---------------- pad x [N,5] -> xp [N,8]; init deg = 1 (self loop) ----------------
__global__ void k_padx_deg(const float* __restrict__ x, float* __restrict__ xp,
                           float* __restrict__ deg, int n) {
  int t = blockIdx.x * 256 + threadIdx.x;
  if (t < n * 8) {
    int i = t >> 3, k = t & 7;
    xp[t] = (k < 5) ? x[i * 5 + k] : 0.f;
  }
  if (t < n) deg[t] = 1.0f;
}

// ---------------- degree count over edges ----------------
__global__ void k_deg(const int* __restrict__ dst, float* __restrict__ deg, int E) {
  int t = blockIdx.x * 256 + threadIdx.x;
  if (t < E) atomicAdd(deg + dst[t], 1.0f);
}

// ---------------- dinv = rsqrt(deg) in place (deg >= 1 always) ----------------
__global__ void k_rsqrt(float* __restrict__ deg, int n) {
  int t = blockIdx.x * 256 + threadIdx.x;
  if (t < n) deg[t] = rsqrtf(deg[t]);
}

// ---------------- WMMA node transform: hs = dinv .* (in @ Wp); acc = hs ----------------
// One wave per 16-node tile; 8 waves / block. KP: padded K (mult of 4);
// NTILES: output 16-col tiles; WROW: weight row width; STORE_F: output row stride.
template <int KP, int NTILES, int IN_STRIDE, int WROW, int STORE_F>
__global__ void k_transform(const float* __restrict__ in, const float* __restrict__ Wp,
                            const float* __restrict__ dinv, float* __restrict__ hs,
                            float* __restrict__ acc, int n) {
  int lane = threadIdx.x & 31;
  int wid = threadIdx.x >> 5;
  int m0 = (blockIdx.x * 8 + wid) * 16;
  if (m0 + 16 > n) return;  // wave-uniform: EXEC stays all-ones for WMMA
  int half = lane >> 4;     // 0: K=k,k+1  1: K=k+2,k+3
  int l15 = lane & 15;
  v8f d[NTILES];
#pragma unroll
  for (int nt = 0; nt < NTILES; ++nt) d[nt] = (v8f){0.f, 0.f, 0.f, 0.f, 0.f, 0.f, 0.f, 0.f};
  int arow = m0 + l15;
#pragma unroll
  for (int kt = 0; kt < KP / 4; ++kt) {
    int k = kt * 4 + half * 2;
    v2f a;
    a.x = in[arow * IN_STRIDE + k];
    a.y = in[arow * IN_STRIDE + k + 1];
#pragma unroll
    for (int nt = 0; nt < NTILES; ++nt) {
      int c = nt * 16 + l15;
      v2f b;
      b.x = Wp[k * WROW + c];
      b.y = Wp[(k + 1) * WROW + c];
      d[nt] = wmma_f32_16x16x4(a, b, d[nt]);
    }
  }
  // D layout: vgpr r, lanes0-15 -> (M=r, N=l15); lanes16-31 -> (M=r+8, N=l15)
#pragma unroll
  for (int r = 0; r < 8; ++r) {
    int row = m0 + r + half * 8;
    float s = dinv[row];
#pragma unroll
    for (int nt = 0; nt < NTILES; ++nt) {
      int c = nt * 16 + l15;
      if (c < STORE_F) {
        float v = d[nt][r] * s;
        hs[row * STORE_F + c] = v;   // pre-scaled features for scatter
        acc[row * STORE_F + c] = v;  // self-loop contribution seeds accumulator
      }
    }
  }
}

// ---------------- edge scatter: acc[dst] += hs[src] (lane-per-feature, coalesced) ------
template <int F, int LOGF>
__global__ void k_scatter(const int* __restrict__ src, const int* __restrict__ dst,
                          const float* __restrict__ hs, float* __restrict__ acc,
                          int total) {
  int t = blockIdx.x * 256 + threadIdx.x;
  if (t >= total) return;
  int e = t >> LOGF;
  int f = t & (F - 1);
  int s = src[e], d = dst[e];
  atomicAdd(acc + d * F + f, hs[s * F + f]);
}

// ---------------- finalize: y = relu(dinv .* acc + b) ----------------
template <int F>
__global__ void k_finalize(const float* __restrict__ acc, const float* __restrict__ dinv,
                           const float* __restrict__ bias, float* __restrict__ y, int n) {
  int t = blockIdx.x * 256 + threadIdx.x;
  if (t >= n * F) return;
  int i = t / F;
  int f = t % F;  // F is a power-of-two constant -> shifts
  float v = dinv[i] * acc[t] + bias[f];
  y[t] = fmaxf(v, 0.f);
}

// ---------------- head GEMM: z[64,32] = H[64,6990] @ Wl1[6990,32] + bl1 ----------------
// H[row][c] = y3[(row*2330 + c/3)*4 + c%3] (y3 stored with padded stride 4).
// One block of 8 waves; wave -> (mtile = wid>>1, ntile = wid&1); K tail guarded.
__global__ void k_head(const float* __restrict__ y3, const float* __restrict__ Wl1,
                       const float* __restrict__ bl1, float* __restrict__ z) {
  int lane = threadIdx.x & 31;
  int wid = threadIdx.x >> 5;
  int mt = wid >> 1;
  int ntb = (wid & 1) * 16;
  int half = lane >> 4, l15 = lane & 15;
  const int K = 2330 * 3;  // 6990
  v8f d = (v8f){0.f, 0.f, 0.f, 0.f, 0.f, 0.f, 0.f, 0.f};
  int arow = mt * 16 + l15;
  for (int kt = 0; kt < (K + 3) / 4; ++kt) {
    int k = kt * 4 + half * 2;
    v2f a, b;
    if (k < K) {
      int node = k / 3, ft = k - node * 3;
      a.x = y3[(arow * 2330 + node) * 4 + ft];
      b.x = Wl1[k * 32 + ntb + l15];
    } else { a.x = 0.f; b.x = 0.f; }
    int k1 = k + 1;
    if (k1 < K) {
      int node = k1 / 3, ft = k1 - node * 3;
      a.y = y3[(arow * 2330 + node) * 4 + ft];
      b.y = Wl1[k1 * 32 + ntb + l15];
    } else { a.y = 0.f; b.y = 0.f; }
    d = wmma_f32_16x16x4(a, b, d);  // full EXEC here; divergence only in loads
  }
#pragma unroll
  for (int r = 0; r < 8; ++r) {
    int row = mt * 16 + r + half * 8;
    int col = ntb + l15;
    z[row * 32 + col] = d[r] + bl1[col];
  }
}

// ---------------- BN (batch dim = 64, biased var) + ReLU + final Linear(32,1) ----------
__global__ void k_bn_out(const float* __restrict__ z, const float* __restrict__ gamma,
                         const float* __restrict__ beta, const float* __restrict__ Wl2,
                         const float* __restrict__ bl2, float* __restrict__ out) {
  __shared__ float sM[32], sS[32], sB[32];
  int t = threadIdx.x;  // 64 threads
  if (t < 32) {
    float s = 0.f, ss = 0.f;
    for (int b = 0; b < 64; ++b) {
      float v = z[b * 32 + t];
      s += v; ss += v * v;
    }
    float mean = s * (1.f / 64.f);
    float var = ss * (1.f / 64.f) - mean * mean;
    sM[t] = mean;
    sS[t] = gamma[t] * rsqrtf(var + 1e-5f);
    sB[t] = beta[t];
  }
  __syncthreads();
  float accv = bl2[0];
  for (int c = 0; c < 32; ++c) {
    float zn = (z[t * 32 + c] - sM[c]) * sS[c] + sB[c];
    accv += fmaxf(zn, 0.f) * Wl2[c];
  }
  out[t] = accv;
}

extern "C" void kernel_launch(void* const* d_in, const int* in_sizes, int n_in,
                              void* d_out, int out_size, void* d_ws, size_t ws_size,
                              hipStream_t stream) {
  const float* x   = (const float*)d_in[0];
  const int*   ei  = (const int*)d_in[1];
  const float* W1  = (const float*)d_in[2];
  const float* b1  = (const float*)d_in[3];
  const float* W2  = (const float*)d_in[4];
  const float* b2  = (const float*)d_in[5];
  const float* W3  = (const float*)d_in[6];
  const float* b3  = (const float*)d_in[7];
  const float* Wl1 = (const float*)d_in[8];
  const float* bl1 = (const float*)d_in[9];
  const float* gm  = (const float*)d_in[10];
  const float* bt  = (const float*)d_in[11];
  const float* Wl2 = (const float*)d_in[12];
  const float* bl2 = (const float*)d_in[13];

  int n = in_sizes[0] / 5;      // 149120 nodes
  int E = in_sizes[1] / 2;      // 2385920 edges
  const int* srcI = ei;
  const int* dstI = ei + E;

  // workspace layout (floats)
  float* ws = (float*)d_ws;
  size_t o = 0;
  float* dinv = ws + o; o += (size_t)n;        // deg -> dinv in place
  float* xp   = ws + o; o += (size_t)n * 8;    // padded input features
  float* hs   = ws + o; o += (size_t)n * 32;   // pre-scaled transformed features
  float* acc  = ws + o; o += (size_t)n * 32;   // aggregation accumulator
  float* ybuf = ws + o; o += (size_t)n * 32;   // layer outputs
  float* w1p  = ws + o; o += 256;
  float* w3p  = ws + o; o += 256;
  float* b3p  = ws + o; o += 16;
  float* z    = ws + o; o += 2048;             // head pre-BN activations [64,32]

  int tilesBlocks = ((n / 16) + 7) / 8;

  k_prep<<<1, 256, 0, stream>>>(W1, W3, b3, w1p, w3p, b3p);
  k_padx_deg<<<(n * 8 + 255) / 256, 256, 0, stream>>>(x, xp, dinv, n);
  k_deg<<<(E + 255) / 256, 256, 0, stream>>>(dstI, dinv, E);
  k_rsqrt<<<(n + 255) / 256, 256, 0, stream>>>(dinv, n);

  // ---- layer 1: 5(->8) -> 32 ----
  k_transform<8, 2, 8, 32, 32><<<tilesBlocks, 256, 0, stream>>>(xp, w1p, dinv, hs, acc, n);
  k_scatter<32, 5><<<(E * 32 + 255) / 256, 256, 0, stream>>>(srcI, dstI, hs, acc, E * 32);
  k_finalize<32><<<(n * 32 + 255) / 256, 256, 0, stream>>>(acc, dinv, b1, ybuf, n);

  // ---- layer 2: 32 -> 16 ----
  k_transform<32, 1, 32, 16, 16><<<tilesBlocks, 256, 0, stream>>>(ybuf, W2, dinv, hs, acc, n);
  k_scatter<16, 4><<<(E * 16 + 255) / 256, 256, 0, stream>>>(srcI, dstI, hs, acc, E * 16);
  k_finalize<16><<<(n * 16 + 255) / 256, 256, 0, stream>>>(acc, dinv, b2, ybuf, n);

  // ---- layer 3: 16 -> 3 (padded to stride 4) ----
  k_transform<16, 1, 16, 16, 4><<<tilesBlocks, 256, 0, stream>>>(ybuf, w3p, dinv, hs, acc, n);
  k_scatter<4, 2><<<(E * 4 + 255) / 256, 256, 0, stream>>>(srcI, dstI, hs, acc, E * 4);
  k_finalize<4><<<(n * 4 + 255) / 256, 256, 0, stream>>>(acc, dinv, b3p, ybuf, n);

  // ---- MLP head ----
  k_head<<<1, 256, 0, stream>>>(ybuf, Wl1, bl1, z);
  k_bn_out<<<1, 64, 0, stream>>>(z, gm, bt, Wl2, bl2, (float*)d_out);
}